// Coulomb_QMMM_10677288698559
// MI455X (gfx1250) — compile-verified
//
#include <hip/hip_runtime.h>
#include <stdint.h>

#define N_QM 256
#define K_EPS 1389.35457644382

typedef float v4f __attribute__((ext_vector_type(4)));
typedef int   v4i __attribute__((ext_vector_type(4)));

// -----------------------------------------------------------------------------
// Zero the 256-entry output (harness poisons d_out; main kernel atomically adds)
// -----------------------------------------------------------------------------
__global__ void esp_zero_kernel(float* __restrict__ out) {
    out[threadIdx.x] = 0.0f;
}

// -----------------------------------------------------------------------------
// Bandwidth-bound streaming kernel.
//  * Param table (256 x 13 floats) staged into LDS via gfx1250
//    GLOBAL_LOAD_ASYNC_TO_LDS_B32 (ASYNCcnt), 13 dwords per thread.
//  * Each thread consumes 4 consecutive edges per iteration so every streaming
//    load is a 16B-aligned b128 with a non-temporal hint (stream > L2 size).
//  * term = mm * ( m/R + dot(d,Rx1)/R^3 + 3*sum(q.*Rx2)/R^5 )
//  * ds_add_f32 into 256 privatized LDS bins, then one global f32 atomic per
//    (block, bin), pre-scaled by K_EPS.
// -----------------------------------------------------------------------------
__global__ __launch_bounds__(256) void esp_main_kernel(
    const float* __restrict__ monos,   // [256]
    const float* __restrict__ dipos,   // [256*3]
    const float* __restrict__ quads,   // [256*9]
    const float* __restrict__ mm,      // [E]
    const float* __restrict__ R1,      // [E]
    const float* __restrict__ Rx1,     // [E*3]
    const float* __restrict__ Rx2,     // [E*9]
    const int*   __restrict__ recv,    // [E]
    float* __restrict__ out,           // [256]
    int E)
{
    __shared__ float s_mono[N_QM];        //  1 KB
    __shared__ float s_dip [3 * N_QM];    //  3 KB
    __shared__ float s_quad[9 * N_QM];    //  9 KB
    __shared__ float s_bins[N_QM];        //  1 KB

    const int t = threadIdx.x;            // 256 threads = 8 wave32
    s_bins[t] = 0.0f;

    // ---- async stage of the parameter table: 13 dwords per thread ----------
    {
        uint32_t o;
        const float* g;

        o = (uint32_t)(uintptr_t)&s_mono[t];
        g = monos + t;
        asm volatile("global_load_async_to_lds_b32 %0, %1, off"
                     :: "v"(o), "v"(g) : "memory");

        #pragma unroll
        for (int k = 0; k < 3; ++k) {
            o = (uint32_t)(uintptr_t)&s_dip[3 * t + k];
            g = dipos + 3 * t + k;
            asm volatile("global_load_async_to_lds_b32 %0, %1, off"
                         :: "v"(o), "v"(g) : "memory");
        }
        #pragma unroll
        for (int k = 0; k < 9; ++k) {
            o = (uint32_t)(uintptr_t)&s_quad[9 * t + k];
            g = quads + 9 * t + k;
            asm volatile("global_load_async_to_lds_b32 %0, %1, off"
                         :: "v"(o), "v"(g) : "memory");
        }
        asm volatile("s_wait_asynccnt 0" ::: "memory");
    }
    __syncthreads();

    const int gid    = (int)(blockIdx.x * blockDim.x) + t;
    const int stride = (int)(gridDim.x * blockDim.x);

    // ---- vectorized main loop: 4 edges per iteration, all loads b128 NT ----
    const int G = E >> 2;                 // number of 4-edge groups
    for (int g4 = gid; g4 < G; g4 += stride) {
        const v4f mm4 = __builtin_nontemporal_load((const v4f*)(mm   + 4 * g4));
        const v4f r14 = __builtin_nontemporal_load((const v4f*)(R1   + 4 * g4));
        const v4i rc4 = __builtin_nontemporal_load((const v4i*)(recv + 4 * g4));

        float xf[12];                     // Rx1 for 4 edges (48B, 16B aligned)
        #pragma unroll
        for (int k = 0; k < 3; ++k)
            ((v4f*)xf)[k] = __builtin_nontemporal_load((const v4f*)(Rx1 + 12 * g4) + k);

        float qf[36];                     // Rx2 for 4 edges (144B, 16B aligned)
        #pragma unroll
        for (int k = 0; k < 9; ++k)
            ((v4f*)qf)[k] = __builtin_nontemporal_load((const v4f*)(Rx2 + 36 * g4) + k);

        #pragma unroll
        for (int j = 0; j < 4; ++j) {
            const float mmv = mm4[j];
            const float r1  = r14[j];
            const int   r   = rc4[j];

            const float b0 = 1.0f / r1;       // B0 = 1/R
            const float i2 = b0 * b0;
            const float b1 = b0 * i2;         // B1 = 1/R^3
            const float b2 = 3.0f * b1 * i2;  // B2 = 3/R^5

            const float dm = s_mono[r];
            const float d1 = s_dip[3 * r + 0] * xf[3 * j + 0]
                           + s_dip[3 * r + 1] * xf[3 * j + 1]
                           + s_dip[3 * r + 2] * xf[3 * j + 2];

            float q2 = 0.0f;
            #pragma unroll
            for (int k = 0; k < 9; ++k)
                q2 += s_quad[9 * r + k] * qf[9 * j + k];

            const float term = mmv * (dm * b0 + d1 * b1 + q2 * b2);
            atomicAdd(&s_bins[r], term);      // ds_add_f32
        }
    }

    // ---- scalar tail (E % 4 edges; empty for E = 4.19M) ---------------------
    for (int e = (G << 2) + gid; e < E; e += stride) {
        const float mmv = mm[e];
        const float r1  = R1[e];
        const int   r   = recv[e];

        const float b0 = 1.0f / r1;
        const float i2 = b0 * b0;
        const float b1 = b0 * i2;
        const float b2 = 3.0f * b1 * i2;

        const float d1 = s_dip[3 * r + 0] * Rx1[3 * e + 0]
                       + s_dip[3 * r + 1] * Rx1[3 * e + 1]
                       + s_dip[3 * r + 2] * Rx1[3 * e + 2];

        float q2 = 0.0f;
        #pragma unroll
        for (int k = 0; k < 9; ++k)
            q2 += s_quad[9 * r + k] * Rx2[9 * e + k];

        atomicAdd(&s_bins[r], mmv * (s_mono[r] * b0 + d1 * b1 + q2 * b2));
    }

    __syncthreads();
    atomicAdd(&out[t], s_bins[t] * (float)K_EPS);  // global_atomic_add_f32
}

// -----------------------------------------------------------------------------
// Inputs (setup_inputs order):
//  0 monos[256,1] 1 dipos[256,3] 2 quads[256,3,3] 3 mm_monos_esp[E,1]
//  4 R1[E,1] 5 Rx1[E,3] 6 Rx2[E,3,3] 7 receivers[E] (int32)
//  8 qm_indices[1,E] (dup of 7) 9 md_mode (0)
// Output: V[256,1] float32
// -----------------------------------------------------------------------------
extern "C" void kernel_launch(void* const* d_in, const int* in_sizes, int n_in,
                              void* d_out, int out_size, void* d_ws, size_t ws_size,
                              hipStream_t stream) {
    const float* monos = (const float*)d_in[0];
    const float* dipos = (const float*)d_in[1];
    const float* quads = (const float*)d_in[2];
    const float* mm    = (const float*)d_in[3];
    const float* R1    = (const float*)d_in[4];
    const float* Rx1   = (const float*)d_in[5];
    const float* Rx2   = (const float*)d_in[6];
    const int*   recv  = (const int*)d_in[7];
    float* out = (float*)d_out;

    const int E = in_sizes[4];   // R1 element count == number of edges

    esp_zero_kernel<<<1, 256, 0, stream>>>(out);

    const int threads = 256;
    int blocks = 2048;           // grid-stride; ~2 groups of 4 edges per thread
    const int G = (E + 3) / 4;
    if (blocks * threads > G) blocks = (G + threads - 1) / threads;
    if (blocks < 1) blocks = 1;

    esp_main_kernel<<<blocks, threads, 0, stream>>>(
        monos, dipos, quads, mm, R1, Rx1, Rx2, recv, out, E);
}